// LucyRNNCell_41034117546123
// MI455X (gfx1250) — compile-verified
//
#include <hip/hip_runtime.h>

// ---------------------------------------------------------------------------
// Fused LucyRNNCell forward for MI455X (gfx1250, wave32, WMMA f32 16x16x4).
// 6 GEMMs (r-branch is dead code in the reference) fused into one kernel:
//   u = LN(x@W_in+b);  z = sig(LN(u@W_z+b));  k = u@W_k+b;  v = u@W_v+b;
//   d = sig(u@W_d+b);  s = d*s_prev + k*v;    c = tanh(LN((u+s)@W_h+b));
//   h = (1-z)*c + z*h_prev
// x-tile DMA'd into LDS by the Tensor Data Mover (TENSORcnt), then the same
// buffer holds u and later u+s; z and k*v live in registers per wave.
// ---------------------------------------------------------------------------

typedef float v2f __attribute__((ext_vector_type(2)));
typedef float v8f __attribute__((ext_vector_type(8)));
typedef unsigned int v4u __attribute__((ext_vector_type(4)));
typedef int v8i __attribute__((ext_vector_type(8)));
typedef int v4i __attribute__((ext_vector_type(4)));

constexpr int   Hd  = 512;     // hidden size == K of every GEMM
constexpr int   BM  = 16;      // rows per workgroup
constexpr int   NT  = 4;       // 16-col WMMA tiles per wave (64 cols/wave * 8 waves = 512)
constexpr int   LDA = 516;     // padded LDS row stride (floats)
constexpr float EPS = 1e-5f;

// TDM-padded LDS layout: 2 pad DWORDs inserted after every 256 DWORDs
// (D# pad_interval=7 -> 256 DWORDs, pad_amount=1 -> 2 DWORDs). Row stride 516.
// Bank map for A-fragment reads (lane=row): ln*516 % 64 = ln*4 -> conflict-free.
__device__ __forceinline__ int lidx(int r, int c) {
  return r * LDA + c + ((c >> 8) << 1);
}

__device__ __forceinline__ float sigmoidf(float x) { return 1.0f / (1.0f + __expf(-x)); }

__device__ __forceinline__ void zero_acc(v8f acc[NT]) {
#pragma unroll
  for (int t = 0; t < NT; ++t)
#pragma unroll
    for (int i = 0; i < 8; ++i) acc[t][i] = 0.0f;
}

// acc(16x64 per wave) += A_lds(16x512) @ W[:, colBase:colBase+64], K=512 step 4.
// A-fragment (16x4 f32): lane = M (=ln), half-wave selects K pair (koff).
// B-fragment (4x16 f32): lane = N (=ln), half-wave selects K pair.
__device__ __forceinline__ void gemm16(const float* __restrict__ A,
                                       const float* __restrict__ W, int colBase,
                                       int ln, int hl, v8f acc[NT]) {
  const int koff = 2 * hl;
  for (int k = 0; k < Hd; k += 4) {
    const float* __restrict__ ap = A + lidx(ln, k + koff);  // pair never straddles pad
    v2f a;
    a.x = ap[0];
    a.y = ap[1];
#pragma unroll
    for (int t = 0; t < NT; ++t) {
      const int c = colBase + 16 * t + ln;
      v2f b;
      b.x = W[(k + koff) * Hd + c];
      b.y = W[(k + koff + 1) * Hd + c];
      // 8 args: (neg_a, A, neg_b, B, c_mod, C, reuse_a, reuse_b)
      acc[t] = __builtin_amdgcn_wmma_f32_16x16x4_f32(
          false, a, false, b, (short)0, acc[t], false, false);
    }
  }
}

__device__ __forceinline__ void add_bias(v8f acc[NT], const float* __restrict__ b,
                                         int colBase, int ln) {
#pragma unroll
  for (int t = 0; t < NT; ++t) {
    const float bc = b[colBase + 16 * t + ln];
#pragma unroll
    for (int i = 0; i < 8; ++i) acc[t][i] += bc;
  }
}

// Row-wise LayerNorm over H=512 directly on the WMMA C layout.
// C layout: acc[t][i] = element(row = i + 8*hl, col = colBase + 16*t + ln).
__device__ __forceinline__ void layernorm16(v8f acc[NT],
                                            const float* __restrict__ g,
                                            const float* __restrict__ bv,
                                            float* s_sum, float* s_ssq,
                                            int colBase, int ln, int hl, int tid) {
  __syncthreads();
  if (tid < BM) { s_sum[tid] = 0.0f; s_ssq[tid] = 0.0f; }
  __syncthreads();

  float p[8], q[8];
#pragma unroll
  for (int i = 0; i < 8; ++i) {
    float s = 0.0f, ss = 0.0f;
#pragma unroll
    for (int t = 0; t < NT; ++t) { const float v = acc[t][i]; s += v; ss += v * v; }
    p[i] = s; q[i] = ss;
  }
  // reduce across the 16 lanes of each half-wave (cols of this wave's slice)
#pragma unroll
  for (int m = 1; m <= 8; m <<= 1) {
#pragma unroll
    for (int i = 0; i < 8; ++i) {
      p[i] += __shfl_xor(p[i], m, 32);
      q[i] += __shfl_xor(q[i], m, 32);
    }
  }
  if (ln == 0) {
#pragma unroll
    for (int i = 0; i < 8; ++i) {
      atomicAdd(&s_sum[i + 8 * hl], p[i]);   // ds_add_f32, cross-wave
      atomicAdd(&s_ssq[i + 8 * hl], q[i]);
    }
  }
  __syncthreads();

  float mu[8], rs[8];
#pragma unroll
  for (int i = 0; i < 8; ++i) {
    const float m1 = s_sum[i + 8 * hl] * (1.0f / Hd);
    const float m2 = s_ssq[i + 8 * hl] * (1.0f / Hd);
    mu[i] = m1;
    rs[i] = rsqrtf(m2 - m1 * m1 + EPS);
  }
#pragma unroll
  for (int t = 0; t < NT; ++t) {
    const int c = colBase + 16 * t + ln;
    const float gc = g[c], bc = bv[c];
#pragma unroll
    for (int i = 0; i < 8; ++i)
      acc[t][i] = (acc[t][i] - mu[i]) * rs[i] * gc + bc;
  }
}

__global__ __launch_bounds__(256) void lucy_cell(
    const float* __restrict__ x, const float* __restrict__ h_prev,
    const float* __restrict__ s_prev,
    const float* __restrict__ W_in, const float* __restrict__ b_in,
    const float* __restrict__ g_ln_in, const float* __restrict__ bl_ln_in,
    const float* __restrict__ g_ln_z, const float* __restrict__ bl_ln_z,
    const float* __restrict__ g_ln_h, const float* __restrict__ bl_ln_h,
    const float* __restrict__ W_z, const float* __restrict__ b_z,
    const float* __restrict__ W_k, const float* __restrict__ b_k,
    const float* __restrict__ W_v, const float* __restrict__ b_v,
    const float* __restrict__ W_d, const float* __restrict__ b_d,
    const float* __restrict__ W_h, const float* __restrict__ b_h,
    float* __restrict__ out_h, float* __restrict__ out_s) {
  __shared__ float sU[BM * LDA];     // x-tile -> u -> u+s   (~33 KB, TDM-padded)
  __shared__ float s_sum[BM], s_ssq[BM];

  const int tid  = threadIdx.x;
  const int wave = tid >> 5;
  const int lane = tid & 31;
  const int ln = lane & 15, hl = lane >> 4;
  const int colBase = wave * (16 * NT);
  const long rowBase = (long)blockIdx.x * BM;

  // ---- Stage x-tile (16x512 f32) into LDS via the Tensor Data Mover -------
  if (wave == 0) {
    // Tensor DMA descriptor (ISA 8.3-8.6): 2D tile, 4B elements, LDS padding
    // of 2 DWORDs per 256 DWORDs => row stride 516 floats (matches lidx()).
    const unsigned long long ga = (unsigned long long)(uintptr_t)(x + rowBase * Hd);
    const unsigned int lds = (unsigned int)(uintptr_t)(void*)sU;
    v4u g0;
    g0[0] = 1u;                                    // count=1, is_restore=0
    g0[1] = lds;                                   // lds_addr (bytes)
    g0[2] = (unsigned int)ga;                      // global_addr[31:0] (tile start)
    g0[3] = (unsigned int)((ga >> 32) & 0x01FFFFFFu) | (2u << 30);  // addr[56:32], type=2
    v8i g1;
    g1[0] = (int)((2u << 16)      // data_size = 4 bytes
                | (1u << 20)      // pad_enable
                | (7u << 22)      // pad_interval: 256 DWORDs
                | (1u << 25));    // pad_amount:   2 DWORDs
    g1[1] = (int)(512u << 16);    // tensor_dim0 = 512   (bits 79:48)
    g1[2] = (int)(32768u << 16);  // tensor_dim1 = 32768 (bits 111:80)
    g1[3] = (int)(512u << 16);    // tile_dim0 = 512     (bits 127:112)
    g1[4] = 16;                   // tile_dim1 = 16 rows (bits 143:128)
    g1[5] = 512;                  // tensor_dim0_stride  (bits 207:160)
    g1[6] = 0;
    g1[7] = 0;                    // tensor_dim1_stride unused for 2D
    v4i g2 = {0, 0, 0, 0};        // 2D tensor: groups 2/3 unused
    v4i g3 = {0, 0, 0, 0};
    v8i g4 = {0, 0, 0, 0, 0, 0, 0, 0};  // extra SGPR group (clang-23 6-arg form)
    __builtin_amdgcn_tensor_load_to_lds(g0, g1, g2, g3, g4, 0);
    __builtin_amdgcn_s_wait_tensorcnt(0);
  }
  __syncthreads();

  v8f acc[NT];

  // ---- Phase 1: u = LN(x @ W_in + b_in) -> sU (overwrites x-tile; safe:
  //      the first barrier inside layernorm16 fences all phase-1 A reads) ----
  zero_acc(acc);
  gemm16(sU, W_in, colBase, ln, hl, acc);
  add_bias(acc, b_in, colBase, ln);
  layernorm16(acc, g_ln_in, bl_ln_in, s_sum, s_ssq, colBase, ln, hl, tid);
#pragma unroll
  for (int t = 0; t < NT; ++t) {
    const int c = colBase + 16 * t + ln;
#pragma unroll
    for (int i = 0; i < 8; ++i) sU[lidx(i + 8 * hl, c)] = acc[t][i];
  }
  __syncthreads();

  // ---- z = sigmoid(LN(u @ W_z + b_z)), kept in registers -------------------
  zero_acc(acc);
  gemm16(sU, W_z, colBase, ln, hl, acc);
  add_bias(acc, b_z, colBase, ln);
  layernorm16(acc, g_ln_z, bl_ln_z, s_sum, s_ssq, colBase, ln, hl, tid);
  v8f zr[NT];
#pragma unroll
  for (int t = 0; t < NT; ++t)
#pragma unroll
    for (int i = 0; i < 8; ++i) zr[t][i] = sigmoidf(acc[t][i]);

  // ---- k = u @ W_k + b_k (registers) ---------------------------------------
  v8f kv[NT];
  zero_acc(kv);
  gemm16(sU, W_k, colBase, ln, hl, kv);
  add_bias(kv, b_k, colBase, ln);

  // ---- v = u @ W_v + b_v;  kv *= v -----------------------------------------
  zero_acc(acc);
  gemm16(sU, W_v, colBase, ln, hl, acc);
  add_bias(acc, b_v, colBase, ln);
#pragma unroll
  for (int t = 0; t < NT; ++t)
#pragma unroll
    for (int i = 0; i < 8; ++i) kv[t][i] *= acc[t][i];

  // ---- decay = sigmoid(u @ W_d + b_d);  s = decay*s_prev + k*v -------------
  zero_acc(acc);
  gemm16(sU, W_d, colBase, ln, hl, acc);
  add_bias(acc, b_d, colBase, ln);
#pragma unroll
  for (int t = 0; t < NT; ++t) {
    const int c = colBase + 16 * t + ln;
#pragma unroll
    for (int i = 0; i < 8; ++i) {
      const long gi = (rowBase + i + 8 * hl) * Hd + c;
      const float s = sigmoidf(acc[t][i]) * s_prev[gi] + kv[t][i];
      out_s[gi] = s;
      kv[t][i] = s;                     // keep s for t = u + s
    }
  }
  __syncthreads();                      // all waves done reading u
#pragma unroll
  for (int t = 0; t < NT; ++t) {
    const int c = colBase + 16 * t + ln;
#pragma unroll
    for (int i = 0; i < 8; ++i) sU[lidx(i + 8 * hl, c)] += kv[t][i];
  }
  __syncthreads();                      // sU now holds u + s

  // ---- c = tanh(LN((u+s) @ W_h + b_h)); h = (1-z)*c + z*h_prev -------------
  zero_acc(acc);
  gemm16(sU, W_h, colBase, ln, hl, acc);
  add_bias(acc, b_h, colBase, ln);
  layernorm16(acc, g_ln_h, bl_ln_h, s_sum, s_ssq, colBase, ln, hl, tid);
#pragma unroll
  for (int t = 0; t < NT; ++t) {
    const int c = colBase + 16 * t + ln;
#pragma unroll
    for (int i = 0; i < 8; ++i) {
      const long gi = (rowBase + i + 8 * hl) * Hd + c;
      const float z = zr[t][i];
      const float cc = tanhf(acc[t][i]);
      out_h[gi] = (1.0f - z) * cc + z * h_prev[gi];
    }
  }
}

extern "C" void kernel_launch(void* const* d_in, const int* in_sizes, int n_in,
                              void* d_out, int out_size, void* d_ws, size_t ws_size,
                              hipStream_t stream) {
  (void)in_sizes; (void)n_in; (void)out_size; (void)d_ws; (void)ws_size;
  // setup_inputs() order:
  // 0:x 1:h_prev 2:s_prev 3:W_in 4:b_in 5:g_ln_in 6:b_ln_in 7:g_ln_r 8:b_ln_r
  // 9:g_ln_z 10:b_ln_z 11:g_ln_h 12:b_ln_h 13:W_r 14:b_r 15:W_z 16:b_z
  // 17:W_k 18:b_k 19:W_v 20:b_v 21:W_h 22:b_h 23:W_d 24:b_d
  const float* x        = (const float*)d_in[0];
  const float* h_prev   = (const float*)d_in[1];
  const float* s_prev   = (const float*)d_in[2];
  const float* W_in     = (const float*)d_in[3];
  const float* b_in     = (const float*)d_in[4];
  const float* g_ln_in  = (const float*)d_in[5];
  const float* bl_ln_in = (const float*)d_in[6];
  const float* g_ln_z   = (const float*)d_in[9];
  const float* bl_ln_z  = (const float*)d_in[10];
  const float* g_ln_h   = (const float*)d_in[11];
  const float* bl_ln_h  = (const float*)d_in[12];
  const float* W_z      = (const float*)d_in[15];
  const float* b_z      = (const float*)d_in[16];
  const float* W_k      = (const float*)d_in[17];
  const float* b_k      = (const float*)d_in[18];
  const float* W_v      = (const float*)d_in[19];
  const float* b_v      = (const float*)d_in[20];
  const float* W_h      = (const float*)d_in[21];
  const float* b_h      = (const float*)d_in[22];
  const float* W_d      = (const float*)d_in[23];
  const float* b_d      = (const float*)d_in[24];

  float* out_h = (float*)d_out;
  float* out_s = out_h + (size_t)32768 * 512;

  dim3 grid(32768 / BM);   // 2048 workgroups
  dim3 block(256);         // 8 waves
  lucy_cell<<<grid, block, 0, stream>>>(
      x, h_prev, s_prev, W_in, b_in, g_ln_in, bl_ln_in,
      g_ln_z, bl_ln_z, g_ln_h, bl_ln_h,
      W_z, b_z, W_k, b_k, W_v, b_v, W_d, b_d, W_h, b_h,
      out_h, out_s);
}